// KnnGrouping_24154896073197
// MI455X (gfx1250) — compile-verified
//
#include <hip/hip_runtime.h>
#include <hip/hip_bf16.h>

typedef __attribute__((ext_vector_type(2))) float v2f;
typedef __attribute__((ext_vector_type(8))) float v8f;

#define BATCH 8
#define CHANS 4
#define NPTS  4096
#define KNN   16

// output layout offsets (floats)
#define O_PTS0 0
#define O_PTS1 (BATCH * NPTS * CHANS)                 // 131072
#define O_GP0  (2 * BATCH * NPTS * CHANS)             // 262144
#define O_GP1  (O_GP0 + BATCH * NPTS * KNN * CHANS)   // 2359296

#define FMAXV 3.402823466e+38f

// -------- kernel 1: pts0, pts1, group_pts0 (pure transpose / replicate) -----
__global__ __launch_bounds__(256) void fill_pts_kernel(
    const float* __restrict__ c0, const float* __restrict__ c1,
    float* __restrict__ out) {
  int g = blockIdx.x * blockDim.x + threadIdx.x;   // 0 .. 32767 (global row)
  if (g >= BATCH * NPTS) return;
  int b = g >> 12;
  int n = g & (NPTS - 1);
  const float* p0 = c0 + (size_t)b * CHANS * NPTS + n;
  const float* p1 = c1 + (size_t)b * CHANS * NPTS + n;
  float4 r0 = make_float4(p0[0], p0[NPTS], p0[2 * NPTS], p0[3 * NPTS]);
  float4 r1 = make_float4(p1[0], p1[NPTS], p1[2 * NPTS], p1[3 * NPTS]);
  ((float4*)(out + O_PTS0))[g] = r0;
  ((float4*)(out + O_PTS1))[g] = r1;
  float4* gp0 = (float4*)(out + O_GP0) + (size_t)g * KNN;
#pragma unroll
  for (int s = 0; s < KNN; ++s) gp0[s] = r0;
}

// -------- kernel 2: fused WMMA distance tiles + top-16 + gather -------------
// One wave (32 lanes) per (batch, 16-query tile). 2048 blocks x 32 threads.
__global__ __launch_bounds__(32) void knn_wmma_kernel(
    const float* __restrict__ c0, const float* __restrict__ c1,
    float* __restrict__ out) {
  __shared__ float dist_tile[16 * 16];   // [row][col] of current 16x16 tile
  __shared__ float hdist[16 * 16];       // per-query heap distances
  __shared__ int   hidx[16 * 16];        // per-query heap db indices (local)

  const int lane = threadIdx.x;          // 0..31 (wave32)
  const int col  = lane & 15;
  const bool hih = lane >= 16;

  const int wg    = blockIdx.x;          // 0..2047
  const int b     = wg >> 8;
  const int qbase = (wg & 255) << 4;

  const float* c0b = c0 + (size_t)b * CHANS * NPTS;
  const float* c1b = c1 + (size_t)b * CHANS * NPTS;

  // ---- A operand: 16x4 fp32, A = -2 * query coords, 4th K component = 0 ----
  // layout: lanes 0-15 -> K=0 (v.x), K=1 (v.y); lanes 16-31 -> K=2 (v.x), K=3 (v.y)
  const int q = qbase + col;
  v2f a;
  if (!hih) {
    a.x = -2.0f * c0b[q];
    a.y = -2.0f * c0b[NPTS + q];
  } else {
    a.x = -2.0f * c0b[2 * NPTS + q];
    a.y = 0.0f;                          // point_dim = 3 -> zero 4th K comp
  }

  // ---- heap init (lane m < 16 owns query row m) ----
  float curmax = FMAXV;
  int   maxpos = 0;
  if (!hih) {
#pragma unroll
    for (int t = 0; t < KNN; ++t) {
      hdist[lane * 16 + t] = FMAXV;
      hidx[lane * 16 + t]  = 0;
    }
  }
  __syncthreads();

  // ---- stream 256 database tiles of 16 points ----
  for (int tile = 0; tile < NPTS / 16; ++tile) {
    const int jbase = tile << 4;
    const int j = jbase + col;

    // db point coords for this lane's column
    const float x0 = c1b[j];
    const float x1 = c1b[NPTS + j];
    const float x2 = c1b[2 * NPTS + j];

    // B operand: 4x16 fp32. lanes 0-15 -> K=0,K=1 rows; lanes 16-31 -> K=2,K=3
    v2f bb;
    if (!hih) { bb.x = x0; bb.y = x1; }
    else      { bb.x = x2; bb.y = 0.0f; }

    // C = ||x_j||^2 splat across all 8 rows this lane holds (column = col)
    const float cn = x0 * x0 + x1 * x1 + x2 * x2;
    v8f cacc = {cn, cn, cn, cn, cn, cn, cn, cn};

    // D[r](lane) = ||x_col||^2 - 2 * y_row . x_col   (row-constant ||y||^2 omitted;
    // it does not change per-query top-k ordering)
    v8f d = __builtin_amdgcn_wmma_f32_16x16x4_f32(
        false, a, false, bb, (short)0, cacc, false, false);

    __syncthreads();   // WAR: previous merge finished reading dist_tile
#pragma unroll
    for (int r = 0; r < 8; ++r) {
      const int row = r + (hih ? 8 : 0);
      dist_tile[row * 16 + col] = d[r];
    }
    __syncthreads();   // RAW: tile staged before merge reads

    // ---- top-16 maintenance: lane m handles query row m ----
    if (!hih) {
      float* hd  = hdist + lane * 16;
      int*   hix = hidx  + lane * 16;
      const float* dr = dist_tile + lane * 16;
      for (int cc = 0; cc < 16; ++cc) {
        const float cand = dr[cc];
        if (cand < curmax) {               // strict <: keep earlier index on ties
          hd[maxpos]  = cand;
          hix[maxpos] = jbase + cc;
          float mx = hd[0]; int mp = 0;
#pragma unroll
          for (int t = 1; t < 16; ++t) {
            const float v = hd[t];
            if (v > mx) { mx = v; mp = t; }
          }
          curmax = mx; maxpos = mp;
        }
      }
    }
  }

  // ---- sort ascending (dist, idx) and write group_pts1 ----
  if (!hih) {
    const float* hd  = hdist + lane * 16;
    const int*   hix = hidx  + lane * 16;
    const int row = b * NPTS + qbase + lane;            // global query row
    float4* gp1 = (float4*)(out + O_GP1) + (size_t)row * KNN;
    unsigned used = 0;
    for (int s = 0; s < KNN; ++s) {
      float best  = FMAXV;
      int   bestj = 0x7fffffff;
      int   bslot = 0;
      for (int t = 0; t < 16; ++t) {
        if (used & (1u << t)) continue;
        const float dv = hd[t];
        const int   jv = hix[t];
        if (dv < best || (dv == best && jv < bestj)) {
          best = dv; bestj = jv; bslot = t;
        }
      }
      used |= 1u << bslot;
      float4 p = make_float4(c1b[bestj],
                             c1b[NPTS + bestj],
                             c1b[2 * NPTS + bestj],
                             c1b[3 * NPTS + bestj]);
      gp1[s] = p;
    }
  }
}

extern "C" void kernel_launch(void* const* d_in, const int* in_sizes, int n_in,
                              void* d_out, int out_size, void* d_ws, size_t ws_size,
                              hipStream_t stream) {
  (void)in_sizes; (void)n_in; (void)out_size; (void)d_ws; (void)ws_size;
  const float* c0 = (const float*)d_in[0];
  const float* c1 = (const float*)d_in[1];
  float* out = (float*)d_out;

  // pts0 / pts1 / group_pts0 (independent of KNN result)
  fill_pts_kernel<<<(BATCH * NPTS + 255) / 256, 256, 0, stream>>>(c0, c1, out);

  // fused WMMA distance + top-16 + gather -> group_pts1
  // one wave per (batch, 16-query tile): 8 * 256 = 2048 blocks of 32 threads
  knn_wmma_kernel<<<BATCH * (NPTS / 16), 32, 0, stream>>>(c0, c1, out);
}